// SuperConv_E_51015621542228
// MI455X (gfx1250) — compile-verified
//
#include <hip/hip_runtime.h>
#include <hip/hip_bf16.h>

#define N_NODES 50000
#define N_EDGES 800000
#define E_TOT   850000   // + self loops
#define IN_CH   128
#define OUT_CH  64
#define HEADS   4
#define HC      256      // HEADS*OUT_CH

typedef __attribute__((ext_vector_type(2))) float v2f;
typedef __attribute__((ext_vector_type(8))) float v8f;

// ---- monotone float<->uint mapping for atomic max on signed floats ----
__device__ __forceinline__ unsigned fflip(float f) {
    unsigned u = __float_as_uint(f);
    return (u & 0x80000000u) ? ~u : (u ^ 0x80000000u);
}
__device__ __forceinline__ float funflip(unsigned u) {
    u = (u & 0x80000000u) ? (u ^ 0x80000000u) : ~u;
    return __uint_as_float(u);
}

__device__ __forceinline__ void edge_pair(const long long* __restrict__ ei, int e,
                                          long long& s, long long& d) {
    if (e < N_EDGES) { s = ei[e]; d = ei[N_EDGES + e]; }
    else             { s = d = (long long)(e - N_EDGES); }   // self loop
}

// ---------------- GEMM: x = feature @ lin_w via V_WMMA_F32_16X16X4_F32 ----------------
// A: [50000,128] row-major, B: [128,256] row-major, X: [50000,256]
// one wave -> one 16x16 f32 tile; K loop of 32 wmma ops (K=4 each)
__global__ __launch_bounds__(256) void gat_gemm_wmma(const float* __restrict__ A,
                                                     const float* __restrict__ B,
                                                     float* __restrict__ X) {
    const int lane  = threadIdx.x & 31;
    const int wave  = threadIdx.x >> 5;            // 0..7
    const int tile  = blockIdx.x * 8 + wave;       // 0..49999
    if (tile >= (N_NODES / 16) * (HC / 16)) return;
    const int tm = tile / (HC / 16);
    const int tn = tile % (HC / 16);
    const int m0 = tm * 16, n0 = tn * 16;

    const int r     = lane & 15;                   // row (A) / col (B,C,D) within tile
    const int khalf = (lane >> 4) * 2;             // 0 for lanes 0-15, 2 for lanes 16-31

    const float* __restrict__ arow = A + (long)(m0 + r) * IN_CH + khalf;
    const float* __restrict__ bcol = B + n0 + r;

    v8f c = {};
    #pragma unroll 4
    for (int k = 0; k < IN_CH; k += 4) {
        v2f a = *(const v2f*)(arow + k);                       // A[m0+r][k+khalf .. +1]
        v2f b;
        b.x = bcol[(long)(k + khalf)     * HC];                // B[k+khalf  ][n0+r]
        b.y = bcol[(long)(k + khalf + 1) * HC];                // B[k+khalf+1][n0+r]
        c = __builtin_amdgcn_wmma_f32_16x16x4_f32(
                /*neg_a=*/false, a, /*neg_b=*/false, b,
                /*c_mod=*/(short)0, c, /*reuse_a=*/false, /*reuse_b=*/false);
    }

    // C/D layout: VGPR v -> M = v (+8 for upper half-wave), N = lane%16
    const int rowBase = m0 + ((lane >> 4) ? 8 : 0);
    float* __restrict__ xp = X + (long)rowBase * HC + n0 + r;
    #pragma unroll
    for (int v = 0; v < 8; ++v) xp[(long)v * HC] = c[v];
}

// ---------------- zero d_out, init amax/denom ----------------
__global__ void gat_init(float* __restrict__ dout, unsigned* __restrict__ amax,
                         float* __restrict__ denom) {
    int i = blockIdx.x * blockDim.x + threadIdx.x;
    if (i < N_NODES * OUT_CH) dout[i] = 0.f;
    if (i < N_NODES * HEADS) { amax[i] = 0u; denom[i] = 0.f; }
}

// ---------------- per-node attention logits ----------------
__global__ void gat_logits(const float* __restrict__ x, const float* __restrict__ att_s,
                           const float* __restrict__ att_d, float* __restrict__ a_src,
                           float* __restrict__ a_dst) {
    int i = blockIdx.x * blockDim.x + threadIdx.x;          // (n,h)
    if (i >= N_NODES * HEADS) return;
    int n = i >> 2, h = i & 3;
    const float* __restrict__ xp = x + (long)n * HC + h * OUT_CH;
    const float* __restrict__ as = att_s + h * OUT_CH;
    const float* __restrict__ ad = att_d + h * OUT_CH;
    float s = 0.f, d = 0.f;
    #pragma unroll
    for (int c = 0; c < OUT_CH; c += 4) {
        float4 xv = *(const float4*)(xp + c);
        float4 sv = *(const float4*)(as + c);
        float4 dv = *(const float4*)(ad + c);
        s += xv.x * sv.x + xv.y * sv.y + xv.z * sv.z + xv.w * sv.w;
        d += xv.x * dv.x + xv.y * dv.y + xv.z * dv.z + xv.w * dv.w;
    }
    a_src[i] = s; a_dst[i] = d;
}

__device__ __forceinline__ float edge_alpha(const float* a_src, const float* a_dst,
                                            long long s, long long d, int h) {
    float z = a_src[s * HEADS + h] + a_dst[d * HEADS + h];
    return z > 0.f ? z : 0.2f * z;                           // leaky_relu(0.2)
}

// ---------------- pass 1: segment max ----------------
__global__ void gat_edge_max(const long long* __restrict__ ei, const float* __restrict__ a_src,
                             const float* __restrict__ a_dst, unsigned* __restrict__ amax) {
    int i = blockIdx.x * blockDim.x + threadIdx.x;          // (e,h)
    if (i >= E_TOT * HEADS) return;
    int e = i >> 2, h = i & 3;
    long long s, d; edge_pair(ei, e, s, d);
    float al = edge_alpha(a_src, a_dst, s, d, h);
    atomicMax(&amax[d * HEADS + h], fflip(al));
}

// ---------------- pass 2: exp + segment sum ----------------
__global__ void gat_edge_expsum(const long long* __restrict__ ei, const float* __restrict__ a_src,
                                const float* __restrict__ a_dst, const unsigned* __restrict__ amax,
                                float* __restrict__ denom, float* __restrict__ wE) {
    int i = blockIdx.x * blockDim.x + threadIdx.x;          // (e,h)
    if (i >= E_TOT * HEADS) return;
    int e = i >> 2, h = i & 3;
    long long s, d; edge_pair(ei, e, s, d);
    float al = edge_alpha(a_src, a_dst, s, d, h);
    float ex = expf(al - funflip(amax[d * HEADS + h]));
    wE[i] = ex;
    atomicAdd(&denom[d * HEADS + h], ex);
}

// ---------------- pass 3: normalize, fold 1/HEADS head-mean ----------------
__global__ void gat_edge_norm(const long long* __restrict__ ei, const float* __restrict__ denom,
                              float* __restrict__ wE) {
    int i = blockIdx.x * blockDim.x + threadIdx.x;          // (e,h)
    if (i >= E_TOT * HEADS) return;
    int e = i >> 2, h = i & 3;
    long long s, d; edge_pair(ei, e, s, d);
    wE[i] = 0.25f * wE[i] / denom[d * HEADS + h];
}

// ---------------- scatter: d_out[dst, c] += w[e,h] * x[src, h*64+c] ----------------
__global__ void gat_scatter(const long long* __restrict__ ei, const float* __restrict__ x,
                            const float* __restrict__ wE, float* __restrict__ dout) {
    long long i = (long long)blockIdx.x * blockDim.x + threadIdx.x;   // (e, j) j in [0,64)
    if (i >= (long long)E_TOT * 64) return;
    int e = (int)(i >> 6);
    int j = (int)(i & 63);
    int h = j >> 4;                      // 4 heads
    int c = (j & 15) * 4;                // 16 float4 chunks of 64 channels
    long long s, d; edge_pair(ei, e, s, d);
    float w = wE[e * HEADS + h];
    float4 xv = *(const float4*)(x + s * HC + h * OUT_CH + c);
    float* __restrict__ op = dout + d * OUT_CH + c;
    atomicAdd(op + 0, w * xv.x);
    atomicAdd(op + 1, w * xv.y);
    atomicAdd(op + 2, w * xv.z);
    atomicAdd(op + 3, w * xv.w);
}

// ---------------- bias ----------------
__global__ void gat_bias(float* __restrict__ dout, const float* __restrict__ bias) {
    int i = blockIdx.x * blockDim.x + threadIdx.x;
    if (i >= N_NODES * OUT_CH) return;
    dout[i] += bias[i & (OUT_CH - 1)];
}

extern "C" void kernel_launch(void* const* d_in, const int* in_sizes, int n_in,
                              void* d_out, int out_size, void* d_ws, size_t ws_size,
                              hipStream_t stream) {
    const float*     feature = (const float*)d_in[0];
    const long long* ei      = (const long long*)d_in[1];   // int64 edge_index [2, E]
    const float*     lin_w   = (const float*)d_in[2];
    const float*     att_src = (const float*)d_in[3];
    const float*     att_dst = (const float*)d_in[4];
    const float*     bias    = (const float*)d_in[5];
    float*           dout    = (float*)d_out;               // [N, 64]

    // workspace layout (floats)
    float*    x     = (float*)d_ws;                         // N*256
    float*    a_src = x + (long)N_NODES * HC;               // N*4
    float*    a_dst = a_src + N_NODES * HEADS;              // N*4
    unsigned* amax  = (unsigned*)(a_dst + N_NODES * HEADS); // N*4
    float*    denom = (float*)(amax + N_NODES * HEADS);     // N*4
    float*    wE    = denom + N_NODES * HEADS;              // E_TOT*4

    const int B = 256;

    // zero output / init softmax state
    gat_init<<<(N_NODES * OUT_CH + B - 1) / B, B, 0, stream>>>(dout, amax, denom);

    // projection GEMM on the f32 matrix pipe (v_wmma_f32_16x16x4_f32)
    const int tiles = (N_NODES / 16) * (HC / 16);           // 50000 waves
    gat_gemm_wmma<<<tiles / 8, B, 0, stream>>>(feature, lin_w, x);

    // node logits
    gat_logits<<<(N_NODES * HEADS + B - 1) / B, B, 0, stream>>>(x, att_src, att_dst, a_src, a_dst);

    // edge passes
    const int eh = E_TOT * HEADS;
    gat_edge_max   <<<(eh + B - 1) / B, B, 0, stream>>>(ei, a_src, a_dst, amax);
    gat_edge_expsum<<<(eh + B - 1) / B, B, 0, stream>>>(ei, a_src, a_dst, amax, denom, wE);
    gat_edge_norm  <<<(eh + B - 1) / B, B, 0, stream>>>(ei, denom, wE);

    // weighted scatter-add into output
    const long long sc = (long long)E_TOT * 64;
    gat_scatter<<<(unsigned)((sc + B - 1) / B), B, 0, stream>>>(ei, x, wE, dout);

    // bias
    gat_bias<<<(N_NODES * OUT_CH + B - 1) / B, B, 0, stream>>>(dout, bias);
}